// SOCA_66168266162879
// MI455X (gfx1250) — compile-verified
//
#include <hip/hip_runtime.h>
#include <math.h>

typedef float v2f __attribute__((ext_vector_type(2)));
typedef float v8f __attribute__((ext_vector_type(8)));
typedef int   v4i __attribute__((ext_vector_type(4)));

#define NB   32
#define HH   64
#define WW   64
#define CC   256
#define CRP  48
#define MM   2304            // 48*48
#define ROWLEN 2304          // xs row length (= M)
#define MATN 256
#define MATSZ (MATN*MATN)    // 65536 elems per batch matrix

// ---- CDNA5 async global->LDS (ASYNCcnt) with safe fallback ------------------
#if defined(__has_builtin)
#  if __has_builtin(__builtin_amdgcn_global_load_async_to_lds_b128)
#    define HAVE_ASYNC_LDS 1
#  endif
#endif
#ifndef HAVE_ASYNC_LDS
#  define HAVE_ASYNC_LDS 0
#endif

__device__ __forceinline__ void async_copy16(const float* g, float* l) {
#if HAVE_ASYNC_LDS
    __builtin_amdgcn_global_load_async_to_lds_b128(
        (__attribute__((address_space(1))) v4i*)(g),
        (__attribute__((address_space(3))) v4i*)(l), 0, 0);
#else
    *(float4*)l = *(const float4*)g;
#endif
}

__device__ __forceinline__ void wait_async() {
#if HAVE_ASYNC_LDS
#  if __has_builtin(__builtin_amdgcn_s_wait_asynccnt)
    __builtin_amdgcn_s_wait_asynccnt(0);
#  else
    asm volatile("s_wait_asynccnt 0x0" ::: "memory");
#  endif
#endif
}

// Map flat crop index t (0..589823) of batch b to index into x (NHWC).
// 4-float chunks (t % 4 == 0) are contiguous and never straddle a crop row
// (boundaries at multiples of 256 and 12288 are both 4-aligned).
__device__ __forceinline__ long long crop_addr(int b, int t) {
    int hp  = t / (CRP * CC);            // 12288
    int rem = t - hp * (CRP * CC);
    int wp  = rem >> 8;                  // /256
    int c   = rem & 255;
    return (((long long)b * HH + 8 + hp) * WW + (8 + wp)) * CC + c;
}

// ---------------- row sums of xs: s[b][i] = sum_k xs[b][i][k] ----------------
__global__ void soca_rowsum(const float* __restrict__ x, float* __restrict__ rowsum) {
    int b = blockIdx.x;
    int i = threadIdx.x;                 // 0..255
    float s = 0.0f;
    int base = i * ROWLEN;
    for (int k = 0; k < ROWLEN; ++k)
        s += x[crop_addr(b, base + k)];
    rowsum[b * MATN + i] = s;
}

// ---------------- Gram + rank-1 correction: cov = (G - s s^T/M)/M ------------
// grid: (256 tiles, NB), block: 32 (one wave per 16x16 tile)
__global__ void soca_gram(const float* __restrict__ x, const float* __restrict__ rowsum,
                          float* __restrict__ cov) {
    int b    = blockIdx.y;
    int tile = blockIdx.x;
    int ti = tile >> 4, tj = tile & 15;
    int lane = threadIdx.x;
    int m  = lane & 15;
    int kh = lane >> 4;

    __shared__ __align__(16) float ldsA[16 * 64];
    __shared__ __align__(16) float ldsB[16 * 64];

    v8f acc = {0.f,0.f,0.f,0.f,0.f,0.f,0.f,0.f};

    for (int k0 = 0; k0 < ROWLEN; k0 += 64) {
        // 256 x b128 chunks per panel
        for (int ch = lane; ch < 256; ch += 32) {
            int r  = ch >> 4;
            int c4 = (ch & 15) << 2;
            async_copy16(x + crop_addr(b, (ti * 16 + r) * ROWLEN + k0 + c4),
                         &ldsA[r * 64 + c4]);
            async_copy16(x + crop_addr(b, (tj * 16 + r) * ROWLEN + k0 + c4),
                         &ldsB[r * 64 + c4]);
        }
        wait_async();
        __syncthreads();
#pragma unroll
        for (int kk = 0; kk < 64; kk += 4) {
            int ka = kk + 2 * kh;
            v2f a, bb;
            a.x  = ldsA[m * 64 + ka];
            a.y  = ldsA[m * 64 + ka + 1];
            bb.x = ldsB[m * 64 + ka];      // B[k][n] = xs[tj*16+n][k], n == lane&15
            bb.y = ldsB[m * 64 + ka + 1];
            acc = __builtin_amdgcn_wmma_f32_16x16x4_f32(false, a, false, bb,
                                                        (short)0, acc, false, false);
        }
        __syncthreads();
    }

    const float Minv = 1.0f / (float)MM;
    int n = m;
    float sj = rowsum[b * MATN + tj * 16 + n];
#pragma unroll
    for (int v = 0; v < 8; ++v) {
        int mi = v + 8 * kh;
        float si = rowsum[b * MATN + ti * 16 + mi];
        float g  = acc[v];
        cov[((long long)b * MATSZ) + (ti * 16 + mi) * MATN + tj * 16 + n] =
            (g - si * sj * Minv) * Minv;
    }
}

// ---------------- trace per batch -------------------------------------------
__global__ void soca_trace(const float* __restrict__ cov, float* __restrict__ norm) {
    int b = blockIdx.x, t = threadIdx.x;
    __shared__ float red[256];
    red[t] = cov[(long long)b * MATSZ + t * (MATN + 1)];
    __syncthreads();
    for (int s = 128; s > 0; s >>= 1) {
        if (t < s) red[t] += red[t + s];
        __syncthreads();
    }
    if (t == 0) norm[b] = red[0];
}

// ---------------- A = cov/norm ; T = 1.5 I - 0.5 A ---------------------------
__global__ void soca_init(const float* __restrict__ cov, const float* __restrict__ norm,
                          float* __restrict__ A, float* __restrict__ T) {
    int b = blockIdx.y;
    int idx = blockIdx.x * 256 + threadIdx.x;        // 0..65535
    float ninv = 1.0f / norm[b];
    long long o = (long long)b * MATSZ + idx;
    float a = cov[o] * ninv;
    A[o] = a;
    int i = idx >> 8, j = idx & 255;
    T[o] = (i == j ? 1.5f : 0.0f) - 0.5f * a;
}

// ---------------- batched GEMM: D = alpha*(P@Q) + beta*I --------------------
// 32x32 tile per wave: 4 accumulators, 4 wmma per fragment-fetch round.
// grid: (64 tiles, NB), block: 32
__global__ void soca_gemm(const float* __restrict__ P, const float* __restrict__ Q,
                          float* __restrict__ D, float alpha, float beta) {
    int b    = blockIdx.y;
    int tile = blockIdx.x;
    int ti = tile >> 3, tj = tile & 7;
    int lane = threadIdx.x;
    int m  = lane & 15;
    int kh = lane >> 4;

    const float* Pb = P + (long long)b * MATSZ;
    const float* Qb = Q + (long long)b * MATSZ;
    float*       Db = D + (long long)b * MATSZ;

    __shared__ __align__(16) float ldsA[32 * 64];   // [row r][k]
    __shared__ __align__(16) float ldsB[64 * 32];   // [k][n]

    v8f acc00 = {0.f,0.f,0.f,0.f,0.f,0.f,0.f,0.f};
    v8f acc01 = {0.f,0.f,0.f,0.f,0.f,0.f,0.f,0.f};
    v8f acc10 = {0.f,0.f,0.f,0.f,0.f,0.f,0.f,0.f};
    v8f acc11 = {0.f,0.f,0.f,0.f,0.f,0.f,0.f,0.f};

    for (int k0 = 0; k0 < MATN; k0 += 64) {
        // A panel: 32 rows x 64 k = 512 b128 chunks
        for (int ch = lane; ch < 512; ch += 32) {
            int r  = ch >> 4;
            int c4 = (ch & 15) << 2;
            async_copy16(Pb + (ti * 32 + r) * MATN + k0 + c4, &ldsA[r * 64 + c4]);
        }
        // B panel: 64 k x 32 n = 512 b128 chunks
        for (int ch = lane; ch < 512; ch += 32) {
            int k  = ch >> 3;
            int n4 = (ch & 7) << 2;
            async_copy16(Qb + (k0 + k) * MATN + tj * 32 + n4, &ldsB[k * 32 + n4]);
        }
        wait_async();
        __syncthreads();
#pragma unroll
        for (int kk = 0; kk < 64; kk += 4) {
            int ka = kk + 2 * kh;
            v2f a0, a1, b0, b1;
            a0.x = ldsA[m * 64 + ka];          a0.y = ldsA[m * 64 + ka + 1];
            a1.x = ldsA[(16 + m) * 64 + ka];   a1.y = ldsA[(16 + m) * 64 + ka + 1];
            b0.x = ldsB[ka * 32 + m];          b0.y = ldsB[(ka + 1) * 32 + m];
            b1.x = ldsB[ka * 32 + 16 + m];     b1.y = ldsB[(ka + 1) * 32 + 16 + m];
            acc00 = __builtin_amdgcn_wmma_f32_16x16x4_f32(false, a0, false, b0,
                                                          (short)0, acc00, false, false);
            acc01 = __builtin_amdgcn_wmma_f32_16x16x4_f32(false, a0, false, b1,
                                                          (short)0, acc01, false, false);
            acc10 = __builtin_amdgcn_wmma_f32_16x16x4_f32(false, a1, false, b0,
                                                          (short)0, acc10, false, false);
            acc11 = __builtin_amdgcn_wmma_f32_16x16x4_f32(false, a1, false, b1,
                                                          (short)0, acc11, false, false);
        }
        __syncthreads();
    }

    int n = m;
#pragma unroll
    for (int v = 0; v < 8; ++v) {
        int gi0 = ti * 32 + v + 8 * kh;
        int gi1 = gi0 + 16;
        int gj0 = tj * 32 + n;
        int gj1 = gj0 + 16;
        Db[gi0 * MATN + gj0] = alpha * acc00[v] + ((gi0 == gj0) ? beta : 0.0f);
        Db[gi0 * MATN + gj1] = alpha * acc01[v] + ((gi0 == gj1) ? beta : 0.0f);
        Db[gi1 * MATN + gj0] = alpha * acc10[v] + ((gi1 == gj0) ? beta : 0.0f);
        Db[gi1 * MATN + gj1] = alpha * acc11[v] + ((gi1 == gj1) ? beta : 0.0f);
    }
}

// ---------------- row mean * sqrt(norm) -------------------------------------
__global__ void soca_rowmean(const float* __restrict__ Yf, const float* __restrict__ norm,
                             float* __restrict__ ymean) {
    int b = blockIdx.x, i = threadIdx.x;
    const float* row = Yf + (long long)b * MATSZ + i * MATN;
    float s = 0.0f;
    for (int j = 0; j < MATN; ++j) s += row[j];
    ymean[b * MATN + i] = s * (1.0f / (float)MATN) * sqrtf(norm[b]);
}

// ---------------- two FC layers (center taps of 3x3 convs) ------------------
__global__ void soca_fc(const float* __restrict__ ymean,
                        const float* __restrict__ w1, const float* __restrict__ b1,
                        const float* __restrict__ w2, const float* __restrict__ b2,
                        float* __restrict__ y2out) {
    int b = blockIdx.x, o = threadIdx.x;
    __shared__ float yin[256];
    __shared__ float y1[256];
    yin[o] = ymean[b * MATN + o];
    __syncthreads();
    const float* w1c = w1 + 4 * MATSZ;     // w1[1][1][i][o], HWIO
    float acc = b1[o];
    for (int i = 0; i < MATN; ++i) acc += yin[i] * w1c[i * MATN + o];
    y1[o] = fmaxf(acc, 0.0f);
    __syncthreads();
    const float* w2c = w2 + 4 * MATSZ;
    float acc2 = b2[o];
    for (int i = 0; i < MATN; ++i) acc2 += y1[i] * w2c[i * MATN + o];
    y2out[b * MATN + o] = 1.0f / (1.0f + expf(-acc2));
}

// ---------------- streaming scale: out = y2[b,c] * x ------------------------
__global__ void soca_scale(const float* __restrict__ x, const float* __restrict__ y2,
                           float* __restrict__ out) {
    long long idx = (long long)blockIdx.x * blockDim.x + threadIdx.x;  // float4 index
    const long long n4 = (long long)NB * HH * WW * CC / 4;             // 8388608
    if (idx >= n4) return;
    long long e  = idx * 4;
    int c  = (int)(e & 255);
    int bb = (int)(e >> 20);                // 64*64*256 = 2^20 elems per batch
    float4 xv = ((const float4*)x)[idx];
    float4 yv = *(const float4*)(y2 + bb * MATN + c);
    float4 o;
    o.x = xv.x * yv.x; o.y = xv.y * yv.y; o.z = xv.z * yv.z; o.w = xv.w * yv.w;
    ((float4*)out)[idx] = o;
}

extern "C" void kernel_launch(void* const* d_in, const int* in_sizes, int n_in,
                              void* d_out, int out_size, void* d_ws, size_t ws_size,
                              hipStream_t stream) {
    const float* x  = (const float*)d_in[0];
    const float* w1 = (const float*)d_in[1];
    const float* b1 = (const float*)d_in[2];
    const float* w2 = (const float*)d_in[3];
    const float* b2 = (const float*)d_in[4];
    float* out = (float*)d_out;

    const long long BUF = (long long)NB * MATSZ;   // 2,097,152 floats = 8 MB
    char* ws = (char*)d_ws;
    float* P0 = (float*)(ws);
    float* P1 = (float*)(ws + BUF * 4 * 1);
    float* P2 = (float*)(ws + BUF * 4 * 2);
    float* P3 = (float*)(ws + BUF * 4 * 3);
    float* P4 = (float*)(ws + BUF * 4 * 4);
    float* rowsum = (float*)(ws + BUF * 4 * 5);           // 32*256
    float* norm   = rowsum + NB * MATN;                   // 32
    float* ymean  = norm + 64;                            // 32*256
    float* y2     = ymean + NB * MATN;                    // 32*256

    // 1) row sums of cropped xs
    soca_rowsum<<<NB, 256, 0, stream>>>(x, rowsum);
    // 2) covariance via WMMA Gram
    soca_gram<<<dim3(256, NB), 32, 0, stream>>>(x, rowsum, P0);
    // 3) trace -> norm
    soca_trace<<<NB, 256, 0, stream>>>(P0, norm);
    // 4) A = cov/norm (P1), T0 = 1.5I - 0.5A (P2)
    soca_init<<<dim3(256, NB), 256, 0, stream>>>(P0, norm, P1, P2);

    auto G = [&](const float* P, const float* Q, float* Dst, float alpha, float beta) {
        soca_gemm<<<dim3(64, NB), 32, 0, stream>>>(P, Q, Dst, alpha, beta);
    };
    // Y0 = A @ T0 ; Z0 = T0 (alias P2)
    G(P1, P2, P3, 1.0f, 0.0f);
    // iter 1: T = 1.5I - 0.5*Z@Y ; Y' = Y@T ; Z' = T@Z
    G(P2, P3, P0, -0.5f, 1.5f);
    G(P3, P0, P1, 1.0f, 0.0f);
    G(P0, P2, P4, 1.0f, 0.0f);     // Y=P1, Z=P4
    // iter 2
    G(P4, P1, P2, -0.5f, 1.5f);
    G(P1, P2, P3, 1.0f, 0.0f);
    G(P2, P4, P0, 1.0f, 0.0f);     // Y=P3, Z=P0
    // iter 3
    G(P0, P3, P1, -0.5f, 1.5f);
    G(P3, P1, P4, 1.0f, 0.0f);
    G(P1, P0, P2, 1.0f, 0.0f);     // Y=P4, Z=P2
    // final: Tf = 3I - Z@Y ; Yf = 0.5 * Y @ Tf
    G(P2, P4, P0, -1.0f, 3.0f);
    G(P4, P0, P1, 0.5f, 0.0f);     // Yf = P1

    // 5) row mean * sqrt(norm)
    soca_rowmean<<<NB, 256, 0, stream>>>(P1, norm, ymean);
    // 6) FC -> relu -> FC -> sigmoid
    soca_fc<<<NB, 256, 0, stream>>>(ymean, w1, b1, w2, b2, y2);
    // 7) broadcast scale (streaming, b128)
    const long long n4 = (long long)NB * HH * WW * CC / 4;
    soca_scale<<<(int)((n4 + 255) / 256), 256, 0, stream>>>(x, y2, out);
}